// QuantumKernelAnsatz_65481071404174
// MI455X (gfx1250) — compile-verified
//
#include <hip/hip_runtime.h>
#include <hip/hip_bf16.h>

// out[b][w] = |cos(x[b][w] + y[b][w])|  for b in [0,1048576), w in [0,4)
//
// Derivation: the reference circuit is a product of single-qubit RY gates,
// RY(x_w) then RY(y_w) on each wire, from |0000>. RY(a)RY(b)=RY(a+b) and
// distinct-wire gates commute, so qubit w = [cos(t/2), sin(t/2)], t=x_w+y_w,
// and <Z_w> = cos^2(t/2)-sin^2(t/2) = cos(t). Output = |cos(x+y)| elementwise.
//
// Memory-bound: 48 MB streamed @ 23.3 TB/s ~= 2.1 us. Use b128 NT loads/stores.

typedef float v4f __attribute__((ext_vector_type(4)));

#define INV_2PI 0.15915494309189535f  // v_cos_f32 input is in revolutions

__global__ __launch_bounds__(256) void qka_abscos_kernel(
    const v4f* __restrict__ x, const v4f* __restrict__ y,
    v4f* __restrict__ out, int n4) {
  int i = blockIdx.x * 256 + threadIdx.x;
  if (i >= n4) return;

  // One 16B row per thread: global_load_b128 with TH=NT (streaming, no reuse)
  v4f a = __builtin_nontemporal_load(x + i);
  v4f b = __builtin_nontemporal_load(y + i);
  v4f t = a + b;

  v4f r;
  r.x = __builtin_fabsf(__builtin_amdgcn_cosf(t.x * INV_2PI));
  r.y = __builtin_fabsf(__builtin_amdgcn_cosf(t.y * INV_2PI));
  r.z = __builtin_fabsf(__builtin_amdgcn_cosf(t.z * INV_2PI));
  r.w = __builtin_fabsf(__builtin_amdgcn_cosf(t.w * INV_2PI));

  __builtin_nontemporal_store(r, out + i);  // global_store_b128 NT
}

extern "C" void kernel_launch(void* const* d_in, const int* in_sizes, int n_in,
                              void* d_out, int out_size, void* d_ws, size_t ws_size,
                              hipStream_t stream) {
  const v4f* x = (const v4f*)d_in[0];   // [B,4] float32, B = 1048576
  const v4f* y = (const v4f*)d_in[1];   // [B,4] float32
  v4f* out = (v4f*)d_out;               // [B,4] float32

  int n4 = in_sizes[0] / 4;             // number of rows (float4 elements)
  int blocks = (n4 + 255) / 256;        // 4096 blocks of 8 wave32
  qka_abscos_kernel<<<blocks, 256, 0, stream>>>(x, y, out, n4);
}